// GatedHebbianUnit_48601849922142
// MI455X (gfx1250) — compile-verified
//
#include <hip/hip_runtime.h>
#include <hip/hip_bf16.h>

// GatedHebbianUnit tick(): out[q] = tanh( sum_{p: dst[p]==q} s[p] * W[p] @ v[src[p]] )
//
// W is 1.07 GB fp32 read exactly once -> memory bound (0.5 FLOP/B, ~46us floor
// at 23.3 TB/s). Strategy: stream W with non-temporal b64 loads directly in the
// V_WMMA_F32_16X16X4_F32 A-operand lane layout, broadcast x from LDS into B
// (replicated across N columns), accumulate 16-row blocks in f32 WMMA, then
// gate + scatter-add via global_atomic_add_f32, with a tanh epilogue pass.
//
// Round 1 -> 2: full K-loop unroll blew past 256 VGPRs (s_set_vgpr_msb churn,
// 3 waves/SIMD). Capped at 256 VGPRs via __launch_bounds__(256,4), unroll 4.
// Round 2 -> 3: unroll 4 drained loadcnt to 0 every body (only 8 loads in
// flight -> ~4-8KB/SIMD, under the ~20KB BW*latency target). Unroll 16 gives
// the scheduler a 32-load window (~64 VGPRs of A in flight, ~130 total) while
// staying under the 256-VGPR / 4-waves-per-SIMD cap.

typedef float v2f __attribute__((ext_vector_type(2)));
typedef float v8f __attribute__((ext_vector_type(8)));

#define N_LAYERS   256
#define LAYER_SIZE 512
#define N_PATHWAYS 1024

__global__ void ghu_zero_kernel(float* __restrict__ out, int n) {
    int i = blockIdx.x * blockDim.x + threadIdx.x;
    if (i < n) out[i] = 0.0f;
}

__global__ void __launch_bounds__(256, 4)
ghu_pathway_matvec_kernel(const float* __restrict__ W,
                          const float* __restrict__ v,
                          const float* __restrict__ s,
                          const int*   __restrict__ dst,
                          const int*   __restrict__ src,
                          float* __restrict__ acc) {
    __shared__ float xs[LAYER_SIZE];

    const int p   = blockIdx.x;       // one workgroup per pathway
    const int tid = threadIdx.x;      // 256 threads = 8 waves (wave32)

    const int   srcRow = src[p];
    const int   dstRow = dst[p];
    const float gate   = s[p];

    // Stage x = v[src[p]] (512 floats) into LDS, coalesced.
    for (int i = tid; i < LAYER_SIZE; i += 256)
        xs[i] = v[(size_t)srcRow * LAYER_SIZE + i];
    __syncthreads();

    const int wave = tid >> 5;          // 0..7
    const int lane = tid & 31;
    const int m    = lane & 15;         // A-matrix row within 16-row block
    const int kOfs = (lane >> 4) << 1;  // 0 (lanes 0-15) or 2 (lanes 16-31)

    const float* Wp = W + (size_t)p * LAYER_SIZE * LAYER_SIZE;

    // Each wave owns 64 rows = 4 blocks of 16 rows.
    for (int blk = 0; blk < 4; ++blk) {
        const int rowBase = wave * 64 + blk * 16;
        // Lane-strided A pointer: lane l streams row (rowBase + l&15),
        // 8B per k-step at byte offset 8*(l>>4) -> lanes l / l+16 cover
        // 16 contiguous bytes; consecutive steps walk the row linearly.
        const float* rowPtr =
            Wp + (size_t)(rowBase + m) * LAYER_SIZE + kOfs;

        v8f c0 = {};
        v8f c1 = {};
#pragma unroll 16
        for (int k = 0; k < LAYER_SIZE; k += 8) {
            // Non-temporal: W is read exactly once, keep it out of L2.
            v2f a0 = __builtin_nontemporal_load((const v2f*)(rowPtr + k));
            v2f a1 = __builtin_nontemporal_load((const v2f*)(rowPtr + k + 4));

            // B (4x16): lane half h supplies K = 2h, 2h+1 -> value depends
            // only on h, so every N column receives the same x values and
            // all 16 columns of D equal y. LDS reads are broadcasts.
            v2f b0, b1;
            b0.x = xs[k + kOfs];
            b0.y = xs[k + kOfs + 1];
            b1.x = xs[k + 4 + kOfs];
            b1.y = xs[k + 4 + kOfs + 1];

            // Two independent accumulator chains keep the XDL pipe busy.
            c0 = __builtin_amdgcn_wmma_f32_16x16x4_f32(
                    false, a0, false, b0, (short)0, c0, false, false);
            c1 = __builtin_amdgcn_wmma_f32_16x16x4_f32(
                    false, a1, false, b1, (short)0, c1, false, false);
        }

        v8f c = c0 + c1;

        // D layout: VGPR r, lane l -> M = r + 8*(l>>4), N = l%16.
        // Take column N==0 (lanes 0 and 16): 8 rows each.
        if ((lane & 15) == 0) {
            const int half = lane >> 4;  // 0 or 1
            float* outBase =
                acc + (size_t)dstRow * LAYER_SIZE + rowBase + half * 8;
#pragma unroll
            for (int r = 0; r < 8; ++r)
                atomicAdd(outBase + r, gate * c[r]);
        }
    }
}

__global__ void ghu_tanh_kernel(float* __restrict__ out, int n) {
    int i = blockIdx.x * blockDim.x + threadIdx.x;
    if (i < n) out[i] = tanhf(out[i]);
}

extern "C" void kernel_launch(void* const* d_in, const int* in_sizes, int n_in,
                              void* d_out, int out_size, void* d_ws, size_t ws_size,
                              hipStream_t stream) {
    const float* W   = (const float*)d_in[0];
    const float* v   = (const float*)d_in[1];
    const float* s   = (const float*)d_in[2];
    const int*   dst = (const int*)d_in[3];
    const int*   src = (const int*)d_in[4];
    float*       out = (float*)d_out;   // [N_LAYERS, LAYER_SIZE] fp32

    const int n = out_size;             // 256 * 512 = 131072

    ghu_zero_kernel<<<(n + 255) / 256, 256, 0, stream>>>(out, n);
    ghu_pathway_matvec_kernel<<<N_PATHWAYS, 256, 0, stream>>>(
        W, v, s, dst, src, out);
    ghu_tanh_kernel<<<(n + 255) / 256, 256, 0, stream>>>(out, n);
}